// LengthRegulator_69063074120137
// MI455X (gfx1250) — compile-verified
//
#include <hip/hip_runtime.h>

#define BB 16
#define SS 512
#define DD 384
#define MAXDUR 8
#define MAXLEN (SS * MAXDUR)      /* 4096 */
#define ROWS_PER_BLOCK 8          /* 8 waves/block, one wave per output row */
#define F4_PER_ROW (DD / 4)       /* 96 x 128-bit chunks per row */

// True vector type so __builtin_nontemporal_store accepts it (HIP float4 is a struct).
typedef float v4f __attribute__((ext_vector_type(4)));

// ---------------------------------------------------------------------------
// Kernel 1: inclusive cumsum of durations per batch (Hillis-Steele in LDS).
// 16 blocks x 512 threads; trivial cost vs the 100 MB copy.
// ---------------------------------------------------------------------------
__global__ __launch_bounds__(SS) void lr_cumsum_kernel(const int* __restrict__ dur,
                                                       int* __restrict__ cs) {
    __shared__ int sm[SS];
    const int b   = blockIdx.x;
    const int tid = threadIdx.x;

    sm[tid] = dur[b * SS + tid];
    __syncthreads();

    #pragma unroll
    for (int off = 1; off < SS; off <<= 1) {
        int v = (tid >= off) ? sm[tid - off] : 0;
        __syncthreads();
        sm[tid] += v;
        __syncthreads();
    }
    cs[b * SS + tid] = sm[tid];
}

// ---------------------------------------------------------------------------
// Kernel 2: expand. One wave (32 lanes) per output row t.
//   - stage this batch's cumsum (2 KB) in LDS
//   - per-wave binary search: first idx with cs[idx] > t  (searchsorted right)
//   - copy 384 floats as 96 x b128 (3 per lane), NT stores so the 100 MB
//     output stream does not evict the 12.6 MB table from L2.
//   - rows past total length: NT zero-fill (reference zeroes invalid frames)
// ---------------------------------------------------------------------------
__global__ __launch_bounds__(256) void lr_expand_kernel(const float* __restrict__ emb,
                                                        const int* __restrict__ cs,
                                                        float* __restrict__ out) {
    __shared__ int scs[SS];

    const int blocks_per_batch = MAXLEN / ROWS_PER_BLOCK;   // 512
    const int b       = blockIdx.x / blocks_per_batch;
    const int rowBase = (blockIdx.x % blocks_per_batch) * ROWS_PER_BLOCK;
    const int tid     = threadIdx.x;

    // Cooperative LDS load of cumsum[b][0..511] (256 threads x 2 entries).
    scs[tid]       = cs[b * SS + tid];
    scs[tid + 256] = cs[b * SS + tid + 256];
    __syncthreads();

    const int total = scs[SS - 1];
    const int wave  = tid >> 5;      // wave32: 8 waves per 256-thread block
    const int lane  = tid & 31;
    const int t     = rowBase + wave;

    v4f* dst = (v4f*)(out + ((size_t)b * MAXLEN + t) * DD);

    if (t >= total) {
        const v4f z = (v4f){0.f, 0.f, 0.f, 0.f};
        #pragma unroll
        for (int c = 0; c < F4_PER_ROW / 32; ++c)
            __builtin_nontemporal_store(z, dst + c * 32 + lane);
        return;
    }

    // Binary search in LDS (wave-uniform: t identical across the wave).
    // searchsorted(cs, t, 'right'): smallest idx with cs[idx] > t.
    // Needs up to ceil(log2(513)) = 10 trips; while-loop keeps lo==hi exact.
    int lo = 0, hi = SS;
    while (lo < hi) {
        int mid = (lo + hi) >> 1;
        if (scs[mid] <= t) lo = mid + 1; else hi = mid;
    }
    if (lo > SS - 1) lo = SS - 1;    // matches reference clip (defensive)

    const v4f* src = (const v4f*)(emb + ((size_t)b * SS + lo) * DD);
    __builtin_prefetch(src + lane, 0, 3);     // global_prefetch_b8 (gfx1250)

    #pragma unroll
    for (int c = 0; c < F4_PER_ROW / 32; ++c) {
        v4f v = src[c * 32 + lane];                           // global_load_b128 (L2-resident)
        __builtin_nontemporal_store(v, dst + c * 32 + lane);  // global_store_b128 th:NT
    }
}

// ---------------------------------------------------------------------------
extern "C" void kernel_launch(void* const* d_in, const int* in_sizes, int n_in,
                              void* d_out, int out_size, void* d_ws, size_t ws_size,
                              hipStream_t stream) {
    const float* emb = (const float*)d_in[0];   // (16, 512, 384) f32
    const int*   dur = (const int*)d_in[1];     // (16, 512) i32
    float*       out = (float*)d_out;           // (16, 4096, 384) f32
    int*         cs  = (int*)d_ws;              // 16*512 ints of scratch

    lr_cumsum_kernel<<<BB, SS, 0, stream>>>(dur, cs);

    const int nblocks = BB * (MAXLEN / ROWS_PER_BLOCK);  // 8192
    lr_expand_kernel<<<nblocks, 256, 0, stream>>>(emb, cs, out);
}